// GINLayer_6957847020296
// MI455X (gfx1250) — compile-verified
//
#include <hip/hip_runtime.h>

typedef float v2f __attribute__((ext_vector_type(2)));
typedef float v8f __attribute__((ext_vector_type(8)));

#define DIM   128
#define PITCH 132          // DIM + 4: float4-aligned, lane->lane bank delta 4
#define BN_EPS 1e-5f

// ---------------------------------------------------------------------------
// CDNA5 async global->LDS copy (ASYNCcnt-tracked, no VGPR round trip).
// VDST operand is the LDS byte address = low 32 bits of the flat pointer
// (LDS aperture has addr[31:0] == LDS offset per ISA 10.2).
// ---------------------------------------------------------------------------
__device__ __forceinline__ void async_copy_b128(void* lds_dst, const void* gsrc) {
    unsigned laddr = (unsigned)(unsigned long long)lds_dst;
    asm volatile("global_load_async_to_lds_b128 %0, %1, off"
                 :: "v"(laddr), "v"(gsrc)
                 : "memory");
}
__device__ __forceinline__ void wait_asynccnt0() {
    asm volatile("s_wait_asynccnt 0" ::: "memory");
}

// ---------------------------------------------------------------------------
// K0: h0 = (1+eps)*x ; also zero the 1024-float stats scratch region
// ---------------------------------------------------------------------------
__global__ void init_h0_kernel(const float* __restrict__ x, const float* __restrict__ eps,
                               float* __restrict__ h0, float* __restrict__ stats, int total4) {
    int i = blockIdx.x * blockDim.x + threadIdx.x;
    if (i < 1024) stats[i] = 0.0f;
    if (i < total4) {
        float s = 1.0f + eps[0];
        float4 v = reinterpret_cast<const float4*>(x)[i];
        v.x *= s; v.y *= s; v.z *= s; v.w *= s;
        reinterpret_cast<float4*>(h0)[i] = v;
    }
}

// ---------------------------------------------------------------------------
// K1: scatter-add. One wave per edge: lane l reads float4 of x[src] (512B
// contiguous per wave) and atomically adds into h0[dst]. x and h0 are both
// L2-resident (51MB each << 192MB L2).
// ---------------------------------------------------------------------------
__global__ void scatter_add_kernel(const float* __restrict__ x, const long long* __restrict__ ei,
                                   float* __restrict__ h0, int E) {
    int t = blockIdx.x * blockDim.x + threadIdx.x;
    int e = t >> 5;
    int l = t & 31;
    if (e < E) {
        long long s = ei[e];                // src row
        long long d = ei[(long long)E + e]; // dst row
        float4 v = reinterpret_cast<const float4*>(x)[s * (DIM / 4) + l];
        float* o = h0 + d * DIM + l * 4;
        atomicAdd(o + 0, v.x);
        atomicAdd(o + 1, v.y);
        atomicAdd(o + 2, v.z);
        atomicAdd(o + 3, v.w);
    }
}

// ---------------------------------------------------------------------------
// K2/K4: out[n,j] = sum_k A'[n,k] * W[j,k] + bias[j]
//   A' = A (GEMM1)  or  relu(scale*A + shift) applied per A-fragment (GEMM2).
// Also accumulates per-column sum / sum-of-squares for the following BatchNorm.
// Block: 256 threads (8 waves) -> 128 rows x 128 cols of output.
// Wave w owns rows [w*16, w*16+16) and all 8 column tiles (A frags reused x8).
// Both LDS tiles are staged with GLOBAL_LOAD_ASYNC_TO_LDS_B128 in identity
// layout; B fragments {W[j,kb], W[j,kb+1]} are adjacent in row-major W, so
// every fragment (A and B) is one conflict-free ds_load_b64.
// ---------------------------------------------------------------------------
template <bool APPLY_BN>
__launch_bounds__(256)
__global__ void gemm_bn_kernel(const float* __restrict__ A, const float* __restrict__ W,
                               const float* __restrict__ bias, float* __restrict__ out,
                               float* __restrict__ stats,
                               const float* __restrict__ inScale, const float* __restrict__ inShift,
                               int N) {
    __shared__ float lA[DIM * PITCH];     // A tile, row-major, padded
    __shared__ float lW[DIM * PITCH];     // W tile, row-major, padded (lW[j*PITCH+k]=W[j,k])
    __shared__ float lSS[2 * DIM];        // BN scale/shift for fused input transform
    __shared__ float lStats[2 * DIM];     // per-column sum, sumsq

    const int tid  = threadIdx.x;
    const int lane = tid & 31;
    const int wv   = tid >> 5;
    const int rowBase = blockIdx.x * 128;

    if (tid < 2 * DIM) lStats[tid] = 0.0f;
    if (APPLY_BN) {
        if (tid < DIM)            lSS[tid] = inScale[tid];
        else if (tid < 2 * DIM)   lSS[tid] = inShift[tid - DIM];
    }

    // Stage W (identity layout): 4096 x 16B async copies, coalesced.
    for (int i = tid; i < DIM * (DIM / 4); i += 256) {
        int j = i >> 5;       // row 0..127
        int c = i & 31;       // float4 chunk 0..31
        async_copy_b128(&lW[j * PITCH + c * 4], &W[j * DIM + c * 4]);
    }

    // Stage 128x128 A tile (identity layout). Wave w stages rows w*16..+15.
    for (int i = 0; i < 16; ++i) {
        int r  = wv * 16 + i;
        int gr = rowBase + r;
        float* lp = &lA[r * PITCH + lane * 4];
        if (gr < N) {
            async_copy_b128(lp, &A[(long long)gr * DIM + lane * 4]);
        } else {
            *reinterpret_cast<float4*>(lp) = make_float4(0.f, 0.f, 0.f, 0.f);
        }
    }
    wait_asynccnt0();
    __syncthreads();

    const int half = lane >> 4;   // K-pair selector per ISA A/B fragment layout
    const int rl   = lane & 15;

    v8f acc[8] = {};
    const float* aRow = &lA[(wv * 16 + rl) * PITCH];
    const float* bRow = &lW[rl * PITCH];    // + ct*16*PITCH + kb (fits DS imm offset)

    #pragma unroll 4
    for (int kk = 0; kk < 32; ++kk) {
        int kb = kk * 4 + 2 * half;
        // A frag: lane(l%16)=row M, vgpr0/1 = K {kb, kb+1}; one ds_load_b64
        v2f a = *reinterpret_cast<const v2f*>(&aRow[kb]);
        if (APPLY_BN) {   // fused BN1 + ReLU on the fragment (broadcast LDS reads)
            v2f sc = *reinterpret_cast<const v2f*>(&lSS[kb]);
            v2f sh = *reinterpret_cast<const v2f*>(&lSS[DIM + kb]);
            a.x = fmaxf(0.f, a.x * sc.x + sh.x);
            a.y = fmaxf(0.f, a.y * sc.y + sh.y);
        }
        #pragma unroll
        for (int ct = 0; ct < 8; ++ct) {
            // B frag: {B[kb,col], B[kb+1,col]} = {W[col,kb], W[col,kb+1]} adjacent
            v2f b = *reinterpret_cast<const v2f*>(&bRow[ct * 16 * PITCH + kb]);
            acc[ct] = __builtin_amdgcn_wmma_f32_16x16x4_f32(
                false, a, false, b, (short)0, acc[ct], false, false);
        }
    }

    // Epilogue: bias add, store pre-BN activations, per-column batch statistics.
    // C layout: vgpr g, lane l -> row = g + 8*(l/16), col = ct*16 + (l%16).
    #pragma unroll
    for (int ct = 0; ct < 8; ++ct) {
        int col = ct * 16 + rl;
        float bcol = bias[col];
        float sv = 0.f, sv2 = 0.f;
        #pragma unroll
        for (int g = 0; g < 8; ++g) {
            int gr = rowBase + wv * 16 + g + 8 * half;
            float v = acc[ct][g] + bcol;
            if (gr < N) {
                out[(long long)gr * DIM + col] = v;
                sv  += v;
                sv2 += v * v;
            }
        }
        atomicAdd(&lStats[col],       sv);   // ds_add_f32, 2-way max conflict
        atomicAdd(&lStats[DIM + col], sv2);
    }
    __syncthreads();
    if (tid < 2 * DIM) atomicAdd(&stats[tid], lStats[tid]);  // 256 global atomics/block
}

// ---------------------------------------------------------------------------
// K3/K5: fold batch stats into per-column scale/shift:
//   scale = gamma * rsqrt(var + eps), shift = beta - mean*scale
// ---------------------------------------------------------------------------
__global__ void bn_params_kernel(const float* __restrict__ stats, const float* __restrict__ gamma,
                                 const float* __restrict__ beta, float* __restrict__ ss, float invN) {
    int c = threadIdx.x;   // 0..127
    float mean = stats[c] * invN;
    float var  = stats[DIM + c] * invN - mean * mean;
    float s    = gamma[c] * rsqrtf(var + BN_EPS);
    ss[c]       = s;
    ss[DIM + c] = beta[c] - mean * s;
}

// ---------------------------------------------------------------------------
// K6: in-place final BN2 + ReLU over d_out, float4-vectorized
// ---------------------------------------------------------------------------
__global__ void bn_relu_kernel(float* __restrict__ out, const float* __restrict__ ss, int total4) {
    int i = blockIdx.x * blockDim.x + threadIdx.x;
    if (i >= total4) return;
    int c4 = i & (DIM / 4 - 1);
    float4 v = reinterpret_cast<float4*>(out)[i];
    float4 s = reinterpret_cast<const float4*>(ss)[c4];
    float4 b = reinterpret_cast<const float4*>(ss + DIM)[c4];
    v.x = fmaxf(0.f, v.x * s.x + b.x);
    v.y = fmaxf(0.f, v.y * s.y + b.y);
    v.z = fmaxf(0.f, v.z * s.z + b.z);
    v.w = fmaxf(0.f, v.w * s.w + b.w);
    reinterpret_cast<float4*>(out)[i] = v;
}

// ---------------------------------------------------------------------------
extern "C" void kernel_launch(void* const* d_in, const int* in_sizes, int n_in,
                              void* d_out, int out_size, void* d_ws, size_t ws_size,
                              hipStream_t stream) {
    const float*      x   = (const float*)d_in[0];
    const long long*  ei  = (const long long*)d_in[1];
    const float*      eps = (const float*)d_in[2];
    const float*      W1  = (const float*)d_in[3];
    const float*      b1  = (const float*)d_in[4];
    const float*      g1  = (const float*)d_in[5];
    const float*      be1 = (const float*)d_in[6];
    const float*      W2  = (const float*)d_in[7];
    const float*      b2  = (const float*)d_in[8];
    const float*      g2  = (const float*)d_in[9];
    const float*      be2 = (const float*)d_in[10];
    float* out = (float*)d_out;

    const int N = in_sizes[0] / DIM;
    const int E = in_sizes[1] / 2;

    float* h0 = (float*)d_ws;                 // [N,128] (1+eps)*x + aggr
    float* t1 = h0 + (size_t)N * DIM;         // [N,128] pre-BN1 activations
    float* sb = t1 + (size_t)N * DIM;         // 1024-float stats block
    float* stats1 = sb;                       // [256] sum, sumsq
    float* ss1    = sb + 256;                 // [256] scale, shift
    float* stats2 = sb + 512;
    float* ss2    = sb + 768;

    const int total4  = N * (DIM / 4);
    const int gblocks = (N + 127) / 128;
    const float invN  = 1.0f / (float)N;

    init_h0_kernel<<<(total4 + 255) / 256, 256, 0, stream>>>(x, eps, h0, sb, total4);
    scatter_add_kernel<<<(E * 32 + 255) / 256, 256, 0, stream>>>(x, ei, h0, E);
    gemm_bn_kernel<false><<<gblocks, 256, 0, stream>>>(h0, W1, b1, t1, stats1,
                                                       nullptr, nullptr, N);
    bn_params_kernel<<<1, DIM, 0, stream>>>(stats1, g1, be1, ss1, invN);
    gemm_bn_kernel<true><<<gblocks, 256, 0, stream>>>(t1, W2, b2, out, stats2,
                                                      ss1, ss1 + DIM, N);
    bn_params_kernel<<<1, DIM, 0, stream>>>(stats2, g2, be2, ss2, invN);
    bn_relu_kernel<<<(total4 + 255) / 256, 256, 0, stream>>>(out, ss2, total4);
}